// Model_12558484374124
// MI455X (gfx1250) — compile-verified
//
#include <hip/hip_runtime.h>
#include <hip/hip_bf16.h>
#include <math.h>

// ---------------------------------------------------------------------------
// KPConv + grid mean pool for MI455X (gfx1250, wave32, WMMA).
//   points1 [65536,3] f32, kernel [25,3] f32, W [25,1,64] f32
//   out: points3 [1000,3] ++ features3 [1000,64]  (67000 f32)
// ---------------------------------------------------------------------------

#define NPTS   65536
#define KNN    10
#define PKER   25
#define PPAD   28        // pad 25 -> 28 (7 chunks of K=4 for wmma 16x16x4)
#define COUT   64
#define NSEG   1000
#define SCH    1024      // support chunk staged in LDS for kNN

typedef float v2f __attribute__((ext_vector_type(2)));
typedef float v4f __attribute__((ext_vector_type(4)));
typedef float v8f __attribute__((ext_vector_type(8)));

__device__ __forceinline__ v8f wmma_16x16x4_f32(v2f a, v2f b, v8f c) {
  // D = A(16x4) * B(4x16) + C(16x16), fp32, wave32.
  return __builtin_amdgcn_wmma_f32_16x16x4_f32(false, a, false, b,
                                               (short)0, c, false, false);
}

// Scan the 8 candidate keys of one D tile into the register-resident top-K.
__device__ __forceinline__ void topk_scan(const v8f& D, int ibase,
                                          float (&key)[KNN], int (&nid)[KNN]) {
  const float dmin =
      fminf(fminf(fminf(D[0], D[1]), fminf(D[2], D[3])),
            fminf(fminf(D[4], D[5]), fminf(D[6], D[7])));
  if (dmin < key[KNN - 1]) {              // single wave-level guard per tile
#pragma unroll
    for (int v = 0; v < 8; ++v) {
      const float kk = D[v];
      if (kk < key[KNN - 1]) {
        key[KNN - 1] = kk;
        nid[KNN - 1] = ibase + v;
#pragma unroll
        for (int s = KNN - 1; s >= 1; --s) {
          if (key[s] < key[s - 1]) {
            float tk = key[s]; key[s] = key[s - 1]; key[s - 1] = tk;
            int   ti = nid[s]; nid[s] = nid[s - 1]; nid[s - 1] = ti;
          }
        }
      }
    }
  }
}

// ------------------------- zero scratch accumulators -----------------------
__global__ void zero_kernel(float* __restrict__ p, int n) {
  int i = blockIdx.x * blockDim.x + threadIdx.x;
  if (i < n) p[i] = 0.0f;
}

// ------------------------------- kNN (WMMA) --------------------------------
// Ranking key per (support m, query n):  |s|^2 + (-2 q) . s
// A (16x4, M=support): lane<16 -> (x,y) of support M=lane ; lane>=16 -> (z, junk)
// B (4x16, N=query):   v0: lanes<16 row K0=-2qx, lanes>=16 row K2=-2qz
//                      v1: lanes<16 row K1=-2qy, lanes>=16 row K3=0   (kills junk)
// C (16x16): row bias |s|^2, lane reads the 8 rows it owns (m0 = 8*(lane>=16)).
__global__ __launch_bounds__(256) void knn_kernel(const float* __restrict__ pts,
                                                  int* __restrict__ out_idx) {
  __shared__ v4f   s_pt[SCH];   // [x, y, z, |s|^2]
  __shared__ float s_sn[SCH];   // compact |s|^2 for C broadcast loads

  const int tid  = threadIdx.x;
  const int lane = tid & 31;
  const int wave = tid >> 5;
  const bool hi  = lane >= 16;
  const int  j   = lane & 15;            // query column within wave tile
  const int  m0  = hi ? 8 : 0;           // row-half handled by this lane
  const int  q   = blockIdx.x * 128 + wave * 16 + j;

  const float qx = pts[q * 3 + 0];
  const float qy = pts[q * 3 + 1];
  const float qz = pts[q * 3 + 2];
  v2f B;
  B.x = hi ? (-2.0f * qz) : (-2.0f * qx);
  B.y = hi ? 0.0f         : (-2.0f * qy);

  float key[KNN];
  int   nid[KNN];
#pragma unroll
  for (int t = 0; t < KNN; ++t) { key[t] = 3.0e38f; nid[t] = 0; }

  const int aoff = hi ? 2 : 0;

  for (int ch = 0; ch < NPTS; ch += SCH) {
    // ---- cooperative stage of support chunk into LDS ----
    for (int i = tid; i < SCH; i += 256) {
      const int p = ch + i;
      const float x = pts[p * 3 + 0];
      const float y = pts[p * 3 + 1];
      const float z = pts[p * 3 + 2];
      const float sn = x * x + y * y + z * z;
      v4f v; v.x = x; v.y = y; v.z = z; v.w = sn;
      s_pt[i] = v;
      s_sn[i] = sn;
    }
    __syncthreads();

    // ---- double-buffered tile loop (2 tiles/iter, no operand rotation) ----
    v2f A0; v4f c00, c01;
    {
      const float* ap = ((const float*)&s_pt[j]) + aoff;
      A0.x = ap[0]; A0.y = ap[1];
      const v4f* cp = (const v4f*)&s_sn[m0];
      c00 = cp[0]; c01 = cp[1];
    }

#pragma unroll 1
    for (int t = 0; t < SCH; t += 32) {
      // buffer 1 <- tile t+16
      v2f A1; v4f c10, c11;
      {
        const float* ap = ((const float*)&s_pt[t + 16 + j]) + aoff;
        A1.x = ap[0]; A1.y = ap[1];
        const v4f* cp = (const v4f*)&s_sn[t + 16 + m0];
        c10 = cp[0]; c11 = cp[1];
      }

      // process tile t (operands loaded one half-iteration ago)
      {
        v8f C;
        C[0] = c00.x; C[1] = c00.y; C[2] = c00.z; C[3] = c00.w;
        C[4] = c01.x; C[5] = c01.y; C[6] = c01.z; C[7] = c01.w;
        const v8f D = wmma_16x16x4_f32(A0, B, C);
        topk_scan(D, ch + t + m0, key, nid);
      }

      // buffer 0 <- tile t+32 (wrap keeps address valid; dead on last iter)
      {
        const int tn = (t + 32) & (SCH - 1);
        const float* ap = ((const float*)&s_pt[tn + j]) + aoff;
        A0.x = ap[0]; A0.y = ap[1];
        const v4f* cp = (const v4f*)&s_sn[tn + m0];
        c00 = cp[0]; c01 = cp[1];
      }

      // process tile t+16
      {
        v8f C;
        C[0] = c10.x; C[1] = c10.y; C[2] = c10.z; C[3] = c10.w;
        C[4] = c11.x; C[5] = c11.y; C[6] = c11.z; C[7] = c11.w;
        const v8f D = wmma_16x16x4_f32(A1, B, C);
        topk_scan(D, ch + t + 16 + m0, key, nid);
      }
    }
    __syncthreads();
  }

  // ---- merge the two lane-halves of each query (lane j gets lane j+16) ----
#pragma unroll
  for (int t = 0; t < KNN; ++t) {
    const float pk = __shfl(key[t], j + 16, 32);
    const int   pi = __shfl(nid[t], j + 16, 32);
    if (!hi && pk < key[KNN - 1]) {
      key[KNN - 1] = pk;
      nid[KNN - 1] = pi;
#pragma unroll
      for (int s = KNN - 1; s >= 1; --s) {
        if (key[s] < key[s - 1]) {
          float tk = key[s]; key[s] = key[s - 1]; key[s - 1] = tk;
          int   ti = nid[s]; nid[s] = nid[s - 1]; nid[s - 1] = ti;
        }
      }
    }
  }
  if (!hi) {
#pragma unroll
    for (int t = 0; t < KNN; ++t) out_idx[q * KNN + t] = nid[t];
  }
}

// -------------------- KPConv influence + WMMA projection -------------------
__global__ __launch_bounds__(256) void kpconv_kernel(const float* __restrict__ pts,
                                                     const float* __restrict__ kern,
                                                     const float* __restrict__ W,
                                                     const int*   __restrict__ nidx,
                                                     float* __restrict__ fsum) {
  __shared__ float sW[PPAD * COUT];     // W zero-padded to 28x64
  __shared__ float sKp[PKER * 3];
  __shared__ float sA[8][32 * PPAD];    // per-wave weighted[q][p] staging
  __shared__ int   sSeg[256];

  const int tid  = threadIdx.x;
  const int lane = tid & 31;
  const int wave = tid >> 5;
  const bool hi  = lane >= 16;
  const int  j   = lane & 15;
  const int  m0  = hi ? 8 : 0;
  const int  q   = blockIdx.x * 256 + tid;

  for (int i = tid; i < PPAD * COUT; i += 256)
    sW[i] = (i < PKER * COUT) ? W[i] : 0.0f;
  for (int i = tid; i < PKER * 3; i += 256) sKp[i] = kern[i];
  __syncthreads();

  const float qx = pts[q * 3 + 0];
  const float qy = pts[q * 3 + 1];
  const float qz = pts[q * 3 + 2];

  // voxel id, matching jnp.floor(p / 0.1f) exactly (true f32 division)
  const int gx = (int)floorf(qx / 0.1f);
  const int gy = (int)floorf(qy / 0.1f);
  const int gz = (int)floorf(qz / 0.1f);
  sSeg[tid] = (gx * 10 + gy) * 10 + gz;

  // weighted[p] = sum_k max(0, 1 - |rel_k - kp_p| / sigma)   (features == 1)
  float wgt[PKER];
#pragma unroll
  for (int p = 0; p < PKER; ++p) wgt[p] = 0.0f;

  for (int k = 0; k < KNN; ++k) {
    const int nb = nidx[q * KNN + k];
    const float rx = pts[nb * 3 + 0] - qx;
    const float ry = pts[nb * 3 + 1] - qy;
    const float rz = pts[nb * 3 + 2] - qz;
#pragma unroll
    for (int p = 0; p < PKER; ++p) {
      const float dx = rx - sKp[p * 3 + 0];
      const float dy = ry - sKp[p * 3 + 1];
      const float dz = rz - sKp[p * 3 + 2];
      const float dist = sqrtf(dx * dx + dy * dy + dz * dz);
      wgt[p] += fmaxf(0.0f, 1.0f - dist / 0.105f);   // sigma = 2.1*0.05
    }
  }

  // stage weighted into A layout source (row-major [qlocal][p], pad p=25..27)
  {
    float* aw = &sA[wave][lane * PPAD];
#pragma unroll
    for (int p = 0; p < PKER; ++p) aw[p] = wgt[p];
    aw[25] = 0.0f; aw[26] = 0.0f; aw[27] = 0.0f;
  }
  __syncthreads();

  // out[16q x 16d] tiles: 7 chained wmma_f32_16x16x4 over the padded P dim,
  // scattered straight into the voxel feature accumulator via f32 atomics.
  for (int mt = 0; mt < 2; ++mt) {
    for (int nt = 0; nt < 4; ++nt) {
      v8f acc = {0.f, 0.f, 0.f, 0.f, 0.f, 0.f, 0.f, 0.f};
#pragma unroll
      for (int c = 0; c < 7; ++c) {
        const float* ap = &sA[wave][(mt * 16 + j) * PPAD + 4 * c + (hi ? 2 : 0)];
        v2f A; A.x = ap[0]; A.y = ap[1];
        const float* bp = &sW[(4 * c + (hi ? 2 : 0)) * COUT + nt * 16 + j];
        v2f Bv; Bv.x = bp[0]; Bv.y = bp[COUT];
        acc = wmma_16x16x4_f32(A, Bv, acc);
      }
#pragma unroll
      for (int v = 0; v < 8; ++v) {
        const int seg = sSeg[wave * 32 + mt * 16 + m0 + v];
        if (seg >= 0 && seg < NSEG)
          atomicAdd(&fsum[seg * COUT + nt * 16 + j], acc[v]);
      }
    }
  }
}

// ----------------------- voxel point-sum / counts --------------------------
__global__ __launch_bounds__(256) void pool_pts_kernel(const float* __restrict__ pts,
                                                       float* __restrict__ psum,
                                                       float* __restrict__ cnt) {
  const int i = blockIdx.x * 256 + threadIdx.x;
  const float x = pts[i * 3 + 0];
  const float y = pts[i * 3 + 1];
  const float z = pts[i * 3 + 2];
  const int gx = (int)floorf(x / 0.1f);
  const int gy = (int)floorf(y / 0.1f);
  const int gz = (int)floorf(z / 0.1f);
  const int seg = (gx * 10 + gy) * 10 + gz;
  if (seg >= 0 && seg < NSEG) {
    atomicAdd(&psum[seg * 3 + 0], x);
    atomicAdd(&psum[seg * 3 + 1], y);
    atomicAdd(&psum[seg * 3 + 2], z);
    atomicAdd(&cnt[seg], 1.0f);
  }
}

// ------------------------------- finalize ----------------------------------
__global__ __launch_bounds__(64) void finalize_kernel(const float* __restrict__ fsum,
                                                      const float* __restrict__ psum,
                                                      const float* __restrict__ cnt,
                                                      float* __restrict__ out) {
  const int s = blockIdx.x;        // 0..999
  const int d = threadIdx.x;       // 0..63
  const float den = fmaxf(cnt[s], 1.0f);
  out[NSEG * 3 + s * COUT + d] = fsum[s * COUT + d] / den;
  if (d < 3) out[s * 3 + d] = psum[s * 3 + d] / den;
}

// ------------------------------- launcher ----------------------------------
extern "C" void kernel_launch(void* const* d_in, const int* in_sizes, int n_in,
                              void* d_out, int out_size, void* d_ws, size_t ws_size,
                              hipStream_t stream) {
  (void)in_sizes; (void)n_in; (void)out_size; (void)ws_size;
  const float* pts  = (const float*)d_in[0];   // [65536,3]
  const float* kern = (const float*)d_in[1];   // [25,3]
  const float* W    = (const float*)d_in[2];   // [25,1,64]
  float* out = (float*)d_out;                  // 3000 + 64000 floats

  char*  ws   = (char*)d_ws;
  int*   nidx = (int*)ws;                                        // N*K ints
  float* fsum = (float*)(ws + (size_t)NPTS * KNN * sizeof(int)); // NSEG*COUT
  float* psum = fsum + NSEG * COUT;                              // NSEG*3
  float* cnt  = psum + NSEG * 3;                                 // NSEG
  const int nzero = NSEG * COUT + NSEG * 3 + NSEG;               // contiguous

  zero_kernel<<<(nzero + 255) / 256, 256, 0, stream>>>(fsum, nzero);
  knn_kernel<<<NPTS / 128, 256, 0, stream>>>(pts, nidx);
  kpconv_kernel<<<NPTS / 256, 256, 0, stream>>>(pts, kern, W, nidx, fsum);
  pool_pts_kernel<<<NPTS / 256, 256, 0, stream>>>(pts, psum, cnt);
  finalize_kernel<<<NSEG, 64, 0, stream>>>(fsum, psum, cnt, out);
}